// NthDistance_89910845375155
// MI455X (gfx1250) — compile-verified
//
#include <hip/hip_runtime.h>
#include <math.h>

typedef __attribute__((ext_vector_type(2))) float v2f;
typedef __attribute__((ext_vector_type(8))) float v8f;

#define BATCH 8
#define NATOM 256
#define NI 128
#define NJ 128
#define NSH 27
#define NQ (NJ*NSH)        /* 3456 */
#define NTOT (NI*NQ)       /* 442368 */
#define NSEL 16
#define INV2SIG2 200.0f    /* 0.5 / (0.05^2) */

#define MT (NI/16)         /* 8   */
#define NT (NQ/16)         /* 216 */
#define NGRP 4             /* N-tiles per wave (4 back-to-back WMMAs) */
#define NTG (NT/NGRP)      /* 54  */
#define TASKS (BATCH*MT*NTG) /* 3456 wave-tasks */

/* workspace layout (in floats) */
#define P_STRIDE (4*NI)                        /* px,py,pz,psq per batch = 512 */
#define Q_OFF    (BATCH*P_STRIDE)
#define Q_STRIDE (4*NQ)                        /* qx,qy,qz,qsq per batch */
#define D_OFF    (Q_OFF + BATCH*Q_STRIDE)
#define DN_OFF   (D_OFF + BATCH*NTOT)          /* + 128 selected values */

// ---------------------------------------------------------------------------
// Kernel 1: cell inverse (adjugate), PBC wrap, build i-point and q-point tables
// ---------------------------------------------------------------------------
__global__ void prep_kernel(const float* __restrict__ pos,
                            const float* __restrict__ cel,
                            const unsigned char* __restrict__ pbc,
                            const int* __restrict__ iidx,
                            const int* __restrict__ jidx,
                            float* __restrict__ ws) {
  const int b = blockIdx.x;
  const int t = threadIdx.x;            // 0..255
  __shared__ float sx[NATOM], sy[NATOM], sz[NATOM];

  const float* C = cel + b * 9;
  const float c00=C[0],c01=C[1],c02=C[2],c10=C[3],c11=C[4],c12=C[5],c20=C[6],c21=C[7],c22=C[8];
  const float det = c00*(c11*c22-c12*c21) - c01*(c10*c22-c12*c20) + c02*(c10*c21-c11*c20);
  const float id  = 1.0f/det;
  const float i00=(c11*c22-c12*c21)*id, i01=(c02*c21-c01*c22)*id, i02=(c01*c12-c02*c11)*id;
  const float i10=(c12*c20-c10*c22)*id, i11=(c00*c22-c02*c20)*id, i12=(c02*c10-c00*c12)*id;
  const float i20=(c10*c21-c11*c20)*id, i21=(c01*c20-c00*c21)*id, i22=(c00*c11-c01*c10)*id;
  const bool p0 = pbc[b*3+0] != 0, p1 = pbc[b*3+1] != 0, p2 = pbc[b*3+2] != 0;

  {   // wrap atom t:  frac = pos * inv(cel);  posw = frac * cel
    const float* P = pos + (b*NATOM + t)*3;
    const float x=P[0], y=P[1], z=P[2];
    float f0 = x*i00 + y*i10 + z*i20;
    float f1 = x*i01 + y*i11 + z*i21;
    float f2 = x*i02 + y*i12 + z*i22;
    if (p0) f0 -= floorf(f0);
    if (p1) f1 -= floorf(f1);
    if (p2) f2 -= floorf(f2);
    sx[t] = f0*c00 + f1*c10 + f2*c20;
    sy[t] = f0*c01 + f1*c11 + f2*c21;
    sz[t] = f0*c02 + f1*c12 + f2*c22;
  }
  __syncthreads();

  float* pb = ws + b * P_STRIDE;
  if (t < NI) {
    const int a = iidx[t];
    const float x=sx[a], y=sy[a], z=sz[a];
    pb[t] = x; pb[NI+t] = y; pb[2*NI+t] = z; pb[3*NI+t] = x*x + y*y + z*z;
  }
  float* qb = ws + Q_OFF + b * Q_STRIDE;
  for (int idx = t; idx < NQ; idx += 256) {
    const int jj = idx / NSH, s = idx % NSH;
    const float fx = (float)(s/9 - 1), fy = (float)((s/3)%3 - 1), fz = (float)(s%3 - 1);
    const float shx = fx*c00 + fy*c10 + fz*c20;
    const float shy = fx*c01 + fy*c11 + fz*c21;
    const float shz = fx*c02 + fy*c12 + fz*c22;
    const int a = jidx[jj];
    const float x = sx[a]+shx, y = sy[a]+shy, z = sz[a]+shz;
    qb[idx] = x; qb[NQ+idx] = y; qb[2*NQ+idx] = z; qb[3*NQ+idx] = x*x + y*y + z*z;
  }
}

// ---------------------------------------------------------------------------
// Kernel 2: all-pairs distances via V_WMMA_F32_16X16X4_F32.
//   A[m][k] = [px,py,pz,1],  B[k][n] = [-2qx,-2qy,-2qz,|q|^2]
//   D = A*B = |q|^2 - 2 p.q ;  dist = sqrt(D + |p|^2)
// One wave32 per (M-tile, group of 4 N-tiles): A + psq loaded once,
// 4 independent WMMAs issued back-to-back. Lane-half operand selection is
// done with address arithmetic / cndmask (no divergent branches).
// ---------------------------------------------------------------------------
__global__ void dist_wmma_kernel(const float* __restrict__ ws,
                                 float* __restrict__ dist) {
  const int wave = threadIdx.x >> 5;
  const int lane = threadIdx.x & 31;
  const int task = blockIdx.x * 8 + wave;     // 0 .. TASKS-1
  const int b  = task / (MT*NTG);
  const int tt = task % (MT*NTG);
  const int mt = tt / NTG;
  const int ng = tt % NTG;
  const int half = lane >> 4;                 // 0: lanes 0-15, 1: lanes 16-31
  const int l15  = lane & 15;

  const float* pb = ws + b * P_STRIDE;
  const float* qb = ws + Q_OFF + b * Q_STRIDE;

  // A (16x4 f32): VGPR0 = K0 (lanes 0-15) / K2 (lanes 16-31); VGPR1 = K1 / K3
  const int mi = mt*16 + l15;
  v2f a;
  a.x = pb[mi + half * (2*NI)];               // x (K0) or z (K2), one load
  const float py = pb[NI + mi];               // unconditional load + cndmask
  a.y = half ? 1.0f : py;                     // 1 (K3) or y (K1)

  // bm.y selects between -2*qy (K1) and qsq (K3): one load + constant scale
  const float yscale = half ? 1.0f : -2.0f;
  const int   boffx  = half * (2*NQ);         // 0 -> qx, 2NQ -> qz
  const int   boffy  = NQ + half * (2*NQ);    // NQ -> qy, 3NQ -> qsq

  // |p|^2 for the 8 output rows this lane owns (C/D: VGPR v -> M = v (+8))
  float psq[8];
  #pragma unroll
  for (int v = 0; v < 8; ++v) psq[v] = pb[3*NI + mt*16 + v + 8*half];

  float* db = dist + b * NTOT;

  #pragma unroll
  for (int g = 0; g < NGRP; ++g) {
    const int nt = ng * NGRP + g;
    const int qi = nt*16 + l15;
    // B (4x16 f32): VGPR0 = K0 / K2 ; VGPR1 = K1 / K3 (N striped across lanes)
    v2f bm;
    bm.x = -2.0f * qb[qi + boffx];
    bm.y = yscale * qb[qi + boffy];

    v8f c = {};
    c = __builtin_amdgcn_wmma_f32_16x16x4_f32(false, a, false, bm,
                                              (short)0, c, false, false);
    #pragma unroll
    for (int v = 0; v < 8; ++v) {
      const int m = mt*16 + v + 8*half;
      const float d2 = c[v] + psq[v];
      db[m * NQ + qi] = __builtin_amdgcn_sqrtf(fmaxf(d2, 0.0f));  // raw v_sqrt_f32
    }
  }
}

// ---------------------------------------------------------------------------
// Kernel 3: exact rank-r selection via bitwise binary search on IEEE bits
// (monotone for non-negative floats). One block per (batch, rank).
// float4 scans -> global_load_b128, L2-resident.
// ---------------------------------------------------------------------------
__global__ void select_kernel(const float* __restrict__ dist,
                              const int* __restrict__ nsel,
                              float* __restrict__ disN) {
  __shared__ unsigned red[256];
  __shared__ unsigned a_sh;
  const int tid = threadIdx.x;
  const int b = blockIdx.x >> 4;
  const unsigned r = (unsigned)nsel[blockIdx.x & 15];
  const float4* d4 = (const float4*)(dist + b * NTOT);
  if (tid == 0) a_sh = 0u;
  __syncthreads();
  for (int bit = 31; bit >= 0; --bit) {
    const unsigned T = a_sh | (1u << bit);
    unsigned cnt = 0;
    for (int idx = tid; idx < NTOT/4; idx += 256) {
      const float4 v = d4[idx];
      cnt += (__float_as_uint(v.x) < T) ? 1u : 0u;
      cnt += (__float_as_uint(v.y) < T) ? 1u : 0u;
      cnt += (__float_as_uint(v.z) < T) ? 1u : 0u;
      cnt += (__float_as_uint(v.w) < T) ? 1u : 0u;
    }
    red[tid] = cnt;
    __syncthreads();
    for (int s = 128; s > 0; s >>= 1) {
      if (tid < s) red[tid] += red[tid + s];
      __syncthreads();
    }
    if (tid == 0 && red[0] <= r) a_sh = T;  // keep max A with count(x < A) <= r
    __syncthreads();
  }
  if (tid == 0) disN[blockIdx.x] = __uint_as_float(a_sh);
}

// ---------------------------------------------------------------------------
// Kernel 4: Gaussian-weighted soft average around each selected distance.
// ---------------------------------------------------------------------------
__global__ void reduce_kernel(const float* __restrict__ dist,
                              const float* __restrict__ disN,
                              float* __restrict__ out) {
  __shared__ float rw[256], rd[256];
  const int tid = threadIdx.x;
  const int b = blockIdx.x >> 4;
  const float dn = disN[blockIdx.x];
  const float4* d4 = (const float4*)(dist + b * NTOT);
  float sw = 0.0f, sd = 0.0f;
  for (int idx = tid; idx < NTOT/4; idx += 256) {
    const float4 v = d4[idx];
    {
      const float df = v.x - dn; const float w = __expf(-df*df*INV2SIG2);
      sw += w; sd += v.x * w;
    }
    {
      const float df = v.y - dn; const float w = __expf(-df*df*INV2SIG2);
      sw += w; sd += v.y * w;
    }
    {
      const float df = v.z - dn; const float w = __expf(-df*df*INV2SIG2);
      sw += w; sd += v.z * w;
    }
    {
      const float df = v.w - dn; const float w = __expf(-df*df*INV2SIG2);
      sw += w; sd += v.w * w;
    }
  }
  rw[tid] = sw; rd[tid] = sd;
  __syncthreads();
  for (int s = 128; s > 0; s >>= 1) {
    if (tid < s) { rw[tid] += rw[tid + s]; rd[tid] += rd[tid + s]; }
    __syncthreads();
  }
  if (tid == 0) out[blockIdx.x] = rd[0] / rw[0];
}

// ---------------------------------------------------------------------------
extern "C" void kernel_launch(void* const* d_in, const int* in_sizes, int n_in,
                              void* d_out, int out_size, void* d_ws, size_t ws_size,
                              hipStream_t stream) {
  const float*         pos  = (const float*)d_in[0];          // (8,256,3)
  const float*         cel  = (const float*)d_in[1];          // (8,3,3)
  const unsigned char* pbc  = (const unsigned char*)d_in[2];  // (8,3) bool
  /* d_in[3] = ent : unused by the reference */
  const int*           iidx = (const int*)d_in[4];            // (128,)
  const int*           jidx = (const int*)d_in[5];            // (128,)
  const int*           nsel = (const int*)d_in[6];            // (16,)
  float* ws   = (float*)d_ws;                                 // ~14.7 MB used
  float* dist = ws + D_OFF;
  float* disN = ws + DN_OFF;
  float* out  = (float*)d_out;                                // (8,16)

  prep_kernel<<<BATCH, 256, 0, stream>>>(pos, cel, pbc, iidx, jidx, ws);
  dist_wmma_kernel<<<TASKS / 8, 256, 0, stream>>>(ws, dist);
  select_kernel<<<BATCH * NSEL, 256, 0, stream>>>(dist, nsel, disN);
  reduce_kernel<<<BATCH * NSEL, 256, 0, stream>>>(dist, disN, out);
}